// Attention_10479720202343
// MI455X (gfx1250) — compile-verified
//
#include <hip/hip_runtime.h>
#include <hip/hip_bf16.h>
#include <math.h>

// ---------------------------------------------------------------------------
// MI455X (gfx1250) fused additive attention.
// B=4, T=256, S=256, D=512.  fp32 end-to-end; GEMM parts use
// V_WMMA_F32_16X16X4_F32 with 2x2 register blocking (32x32 C per wave).
// Score core (tanh-bound) uses float4-vectorized fused VALU/TRANS.
// ---------------------------------------------------------------------------

typedef __attribute__((ext_vector_type(2))) float v2f;
typedef __attribute__((ext_vector_type(8))) float v8f;

#define Bb 4
#define Tt 256
#define Ss 256
#define Dd 512

// ---------------------------------------------------------------------------
// WMMA f32 2x2-blocked tile K-loop: one wave accumulates a 32x32 C block.
//  A: row-major, pre-offset to tile row 0 (32 rows, lda stride)
//  B: row-major, pre-offset to tile col 0 (32 cols, ldb stride)
//  Per k-step: 2 A fragments + 2 B fragments feed 4 WMMAs (4 loads : 4 wmma).
//  Fragment layout per CDNA5 ISA 7.12.2:
//   A 16x4 f32 : lanes 0-15 hold {K=0,K=1}, lanes 16-31 hold {K=2,K=3}, M=lane%16
//   B 4x16 f32 : mirrored, N=lane%16
//   C/D        : VGPR r -> M = r + 8*(lane/16), N = lane%16
// ---------------------------------------------------------------------------
__device__ inline void wmma_k_loop_2x2(const float* __restrict__ A, int lda,
                                       const float* __restrict__ B, int ldb,
                                       int K, int lr, int half,
                                       v8f& c00, v8f& c01, v8f& c10, v8f& c11) {
#pragma unroll 2
  for (int k = 0; k < K; k += 4) {
    const int ka = k + half * 2;
    v2f a0, a1, b0, b1;
    a0.x = A[lr * lda + ka];
    a0.y = A[lr * lda + ka + 1];
    a1.x = A[(lr + 16) * lda + ka];
    a1.y = A[(lr + 16) * lda + ka + 1];
    b0.x = B[ka * ldb + lr];
    b0.y = B[(ka + 1) * ldb + lr];
    b1.x = B[ka * ldb + 16 + lr];
    b1.y = B[(ka + 1) * ldb + 16 + lr];
    c00 = __builtin_amdgcn_wmma_f32_16x16x4_f32(false, a0, false, b0, (short)0, c00, false, false);
    c01 = __builtin_amdgcn_wmma_f32_16x16x4_f32(false, a0, false, b1, (short)0, c01, false, false);
    c10 = __builtin_amdgcn_wmma_f32_16x16x4_f32(false, a1, false, b0, (short)0, c10, false, false);
    c11 = __builtin_amdgcn_wmma_f32_16x16x4_f32(false, a1, false, b1, (short)0, c11, false, false);
  }
}

// Store a 32x32 accumulator block (optionally with per-column bias).
__device__ inline void store_2x2(float* __restrict__ Y, int ldy,
                                 int m0, int n0, int lr, int half,
                                 const v8f& c00, const v8f& c01,
                                 const v8f& c10, const v8f& c11,
                                 float bias0, float bias1) {
#pragma unroll
  for (int r = 0; r < 8; ++r) {
    const int row0 = m0 + half * 8 + r;
    const int row1 = row0 + 16;
    Y[(size_t)row0 * ldy + n0 + lr]      = c00[r] + bias0;
    Y[(size_t)row0 * ldy + n0 + 16 + lr] = c01[r] + bias1;
    Y[(size_t)row1 * ldy + n0 + lr]      = c10[r] + bias0;
    Y[(size_t)row1 * ldy + n0 + 16 + lr] = c11[r] + bias1;
  }
}

// ---------------------------------------------------------------------------
// Kernel 1: wq = output @ Wq + bq   (z==0)
//           uh = context @ Wc       (z==1)
// grid (8, 16, 2), block (32, 4): each wave computes a 32x32 tile of (1024x512).
// ---------------------------------------------------------------------------
__global__ void proj_kernel(const float* __restrict__ out_states,
                            const float* __restrict__ ctx_states,
                            const float* __restrict__ Wq,
                            const float* __restrict__ bq,
                            const float* __restrict__ Wc,
                            float* __restrict__ wq,
                            float* __restrict__ uh) {
  const int lane = threadIdx.x;
  const int lr = lane & 15;
  const int half = lane >> 4;
  const int m0 = (blockIdx.x * 4 + threadIdx.y) * 32;   // row tile in [0,1024)
  const int n0 = blockIdx.y * 32;                       // col tile in [0,512)
  const bool is_q = (blockIdx.z == 0);

  const float* X = is_q ? out_states : ctx_states;
  const float* W = is_q ? Wq : Wc;
  float* Y = is_q ? wq : uh;

  v8f c00 = {}, c01 = {}, c10 = {}, c11 = {};
  wmma_k_loop_2x2(X + (size_t)m0 * Dd, Dd, W + n0, Dd, Dd, lr, half,
                  c00, c01, c10, c11);

  const float bias0 = is_q ? bq[n0 + lr] : 0.0f;
  const float bias1 = is_q ? bq[n0 + 16 + lr] : 0.0f;
  store_2x2(Y, Dd, m0, n0, lr, half, c00, c01, c10, c11, bias0, bias1);
}

// ---------------------------------------------------------------------------
// Kernel 2: fused score + masked softmax.
//   score[b,t,s] = sum_d v[d] * tanh(wq[b,t,d] + uh[b,s,d])
//   attn = softmax with: rowmax over ALL s, exp, *(1-mask), renormalize
// grid (T, B), block 256 (8 waves). wq row + v in LDS; uh streams from L2 as
// b128 loads (lane l owns contiguous float4 at d = d0 + 4l).
// ---------------------------------------------------------------------------
__global__ void score_softmax_kernel(const float* __restrict__ wq,
                                     const float* __restrict__ uh,
                                     const int* __restrict__ mask,
                                     const float* __restrict__ v,
                                     float* __restrict__ attn) {
  __shared__ __align__(16) float s_wq[Dd];
  __shared__ __align__(16) float s_v[Dd];
  __shared__ float s_score[Ss];
  __shared__ float s_red[8];

  const int t = blockIdx.x;
  const int b = blockIdx.y;
  const int tid = threadIdx.x;
  const int lane = tid & 31;
  const int wave = tid >> 5;

  const size_t row_bt = (size_t)(b * Tt + t);
  {
    // 256 threads x float2 covers 512 floats in one shot
    const float2 wq2 = ((const float2*)(wq + row_bt * Dd))[tid];
    const float2 v2 = ((const float2*)v)[tid];
    ((float2*)s_wq)[tid] = wq2;
    ((float2*)s_v)[tid] = v2;
  }
  __syncthreads();

  // each wave sweeps s; lanes parallelize d as contiguous float4 chunks
  for (int s = wave; s < Ss; s += 8) {
    const float* uhrow = uh + ((size_t)b * Ss + s) * Dd;
    float acc = 0.0f;
#pragma unroll
    for (int d0 = 0; d0 < Dd; d0 += 128) {
      const int d = d0 + lane * 4;
      const float4 u = *(const float4*)(uhrow + d);
      const float4 w = *(const float4*)(s_wq + d);
      const float4 vv = *(const float4*)(s_v + d);
      acc = fmaf(vv.x, tanhf(w.x + u.x), acc);
      acc = fmaf(vv.y, tanhf(w.y + u.y), acc);
      acc = fmaf(vv.z, tanhf(w.z + u.z), acc);
      acc = fmaf(vv.w, tanhf(w.w + u.w), acc);
    }
#pragma unroll
    for (int off = 16; off >= 1; off >>= 1) acc += __shfl_xor(acc, off, 32);
    if (lane == 0) s_score[s] = acc;
  }
  __syncthreads();

  // block-wide max over all 256 scores (pre-mask, per reference)
  const float my = s_score[tid];
  float mx = my;
#pragma unroll
  for (int off = 16; off >= 1; off >>= 1) mx = fmaxf(mx, __shfl_xor(mx, off, 32));
  if (lane == 0) s_red[wave] = mx;
  __syncthreads();
  float rowmax = s_red[0];
#pragma unroll
  for (int i = 1; i < 8; ++i) rowmax = fmaxf(rowmax, s_red[i]);
  __syncthreads();

  const float keep = 1.0f - (float)mask[b * Ss + tid];
  const float e = __expf(my - rowmax) * keep;

  float sm = e;
#pragma unroll
  for (int off = 16; off >= 1; off >>= 1) sm += __shfl_xor(sm, off, 32);
  if (lane == 0) s_red[wave] = sm;
  __syncthreads();
  float denom = 0.0f;
#pragma unroll
  for (int i = 0; i < 8; ++i) denom += s_red[i];

  attn[row_bt * Ss + tid] = e / denom;
}

// ---------------------------------------------------------------------------
// Kernel 3: mix[b] = attn[b] @ context[b]   ((T x S) @ (S x D)), K=256
// grid (2, 16, B), block (32, 4): 32x32 WMMA tile per wave.
// ---------------------------------------------------------------------------
__global__ void mix_kernel(const float* __restrict__ attn,
                           const float* __restrict__ ctx_states,
                           float* __restrict__ mix) {
  const int lane = threadIdx.x;
  const int lr = lane & 15;
  const int half = lane >> 4;
  const int b = blockIdx.z;
  const int m0 = (blockIdx.x * 4 + threadIdx.y) * 32;   // t tile in [0,256)
  const int n0 = blockIdx.y * 32;                       // d tile in [0,512)

  const float* A = attn + (size_t)b * Tt * Ss;          // lda = S
  const float* Bm = ctx_states + (size_t)b * Ss * Dd;   // ldb = D

  v8f c00 = {}, c01 = {}, c10 = {}, c11 = {};
  wmma_k_loop_2x2(A + (size_t)m0 * Ss, Ss, Bm + n0, Dd, Ss, lr, half,
                  c00, c01, c10, c11);

  store_2x2(mix + (size_t)b * Tt * Dd, Dd, m0, n0, lr, half,
            c00, c01, c10, c11, 0.0f, 0.0f);
}

// ---------------------------------------------------------------------------
// Kernel 4: out = mix @ Wout[0:D,:] + output @ Wout[D:2D,:] + bout
// grid (8, 16), block (32, 4).  Two fused K=512 WMMA loops, one accumulator.
// ---------------------------------------------------------------------------
__global__ void out_kernel(const float* __restrict__ mix,
                           const float* __restrict__ out_states,
                           const float* __restrict__ Wout,
                           const float* __restrict__ bout,
                           float* __restrict__ out) {
  const int lane = threadIdx.x;
  const int lr = lane & 15;
  const int half = lane >> 4;
  const int m0 = (blockIdx.x * 4 + threadIdx.y) * 32;   // row tile in [0,1024)
  const int n0 = blockIdx.y * 32;                       // col tile in [0,512)

  v8f c00 = {}, c01 = {}, c10 = {}, c11 = {};
  // combined[:, 0:512]   = mix    vs Wout rows [0,512)
  wmma_k_loop_2x2(mix + (size_t)m0 * Dd, Dd, Wout + n0, Dd, Dd, lr, half,
                  c00, c01, c10, c11);
  // combined[:, 512:1024] = output vs Wout rows [512,1024)
  wmma_k_loop_2x2(out_states + (size_t)m0 * Dd, Dd,
                  Wout + (size_t)Dd * Dd + n0, Dd, Dd, lr, half,
                  c00, c01, c10, c11);

  store_2x2(out, Dd, m0, n0, lr, half, c00, c01, c10, c11,
            bout[n0 + lr], bout[n0 + 16 + lr]);
}

// ---------------------------------------------------------------------------
// Launch
// ---------------------------------------------------------------------------
extern "C" void kernel_launch(void* const* d_in, const int* in_sizes, int n_in,
                              void* d_out, int out_size, void* d_ws, size_t ws_size,
                              hipStream_t stream) {
  const float* out_states = (const float*)d_in[0];  // (B,T,D)
  const float* ctx_states = (const float*)d_in[1];  // (B,S,D)
  const int* mask = (const int*)d_in[2];            // (B,S)
  const float* Wq = (const float*)d_in[3];          // (D,D)
  const float* bq = (const float*)d_in[4];          // (D)
  const float* Wc = (const float*)d_in[5];          // (D,D)
  const float* v = (const float*)d_in[6];           // (D)
  const float* Wout = (const float*)d_in[7];        // (2D,D)
  const float* bout = (const float*)d_in[8];        // (D)

  float* out = (float*)d_out;                           // (B,T,D) = 524288 floats
  float* attn = (float*)d_out + (size_t)Bb * Tt * Dd;   // (B,T,S) = 262144 floats

  float* wq = (float*)d_ws;                             // (B*T, D)  2 MB
  float* uh = wq + (size_t)Bb * Tt * Dd;                // (B*S, D)  2 MB
  float* mix = uh + (size_t)Bb * Ss * Dd;               // (B*T, D)  2 MB

  // 1) projections (WMMA f32, 2x2 blocked)
  proj_kernel<<<dim3(8, 16, 2), dim3(32, 4), 0, stream>>>(
      out_states, ctx_states, Wq, bq, Wc, wq, uh);

  // 2) fused score + masked softmax (VALU/TRANS bound, b128 streams)
  score_softmax_kernel<<<dim3(Tt, Bb), dim3(256), 0, stream>>>(
      wq, uh, mask, v, attn);

  // 3) mix = attn @ context (WMMA f32, 2x2 blocked)
  mix_kernel<<<dim3(2, 16, Bb), dim3(32, 4), 0, stream>>>(
      attn, ctx_states, mix);

  // 4) out = [mix, output] @ Wout + bout (WMMA f32, 2x2 blocked)
  out_kernel<<<dim3(8, 16), dim3(32, 4), 0, stream>>>(
      mix, out_states, Wout, bout, out);
}